// LSTM_86157043958127
// MI455X (gfx1250) — compile-verified
//
#include <hip/hip_runtime.h>
#include <hip/hip_bf16.h>
#include <math.h>

// ---------------------------------------------------------------------------
// 2-layer LSTM single step (batch=1, seq=1) + linear head, fused into ONE
// persistent workgroup.  Matvecs run on the CDNA5 matrix pipe via
// V_WMMA_F32_16X16X4_F32 (fp32 in/out).  Each wave owns exactly two 16-row
// gate tiles and drives FOUR independent WMMA accumulator chains
// (2 tiles x 2 K-parities).  Weight loads are explicitly batched (6 chunks =
// 12 loads in flight per wave) and all weight matrices are prefetched into
// the cache hierarchy (global_prefetch_b8) during the staging phase.
// ---------------------------------------------------------------------------

#define LAG 100            // input size
#define HID 150            // hidden size
#define G4  (4 * HID)      // 600 gate rows per layer
#define NTILES 38          // ceil(600 / 16)
#define NWAVES 19          // 38 tiles = 2 per wave, exactly
#define NTHREADS (NWAVES * 32)
#define BATCH 6            // K-chunks staged per load batch

typedef __attribute__((ext_vector_type(2))) float v2f;
typedef __attribute__((ext_vector_type(8))) float v8f;

__device__ __forceinline__ float sigmoidf_(float x) {
    return 1.0f / (1.0f + expf(-x));
}

__device__ __forceinline__ v8f wmma4(v2f A, v2f B, v8f C) {
    return __builtin_amdgcn_wmma_f32_16x16x4_f32(false, A, false, B,
                                                 (short)0, C, false, false);
}

// Dual-tile matvec accumulate with batched loads:
//   a00/a01 : K-parity accumulator pair for tile rows row0..row0+15
//   a10/a11 : K-parity accumulator pair for tile rows row1..row1+15
//   vec     : LDS vector, zero-padded past K (no tail guards needed)
//   W       : row-major [600 x ldW] global weights; row0/row1 pre-clamped+n
// Gate value for row r lands in (aX0[0] + aX1[0]) on lanes 0..15.
template <int K>
__device__ __forceinline__ void wmma_matvec2(v8f& a00, v8f& a01,
                                             v8f& a10, v8f& a11,
                                             const float* __restrict__ W, int ldW,
                                             const float* vec,
                                             int row0, int row1, int lane) {
    const int  n   = lane & 15;
    const bool hi  = (lane >= 16);
    const float* w0 = W + row0 * ldW;
    const float* w1 = W + row1 * ldW;
    constexpr int NCH = (K + 3) / 4;        // K-chunks (25 or 38)

#pragma unroll
    for (int cb = 0; cb < NCH; cb += BATCH) {
        v2f B0b[BATCH], B1b[BATCH], Ab[BATCH];

        // --- stage: issue all loads of this batch back-to-back -------------
#pragma unroll
        for (int j = 0; j < BATCH; ++j) {
            const int c = cb + j;
            if (c < NCH) {
                const int k = c * 4;
                // B 4x16: VGPR0 = {K=k | K=k+2}, VGPR1 = {K=k+1 | K=k+3}
                int colb = hi ? (k + 2) : k;
                if (colb > K - 2) colb = K - 2;   // K%4==2 tail (A pad is 0)
                B0b[j].x = w0[colb];
                B0b[j].y = w0[colb + 1];
                B1b[j].x = w1[colb];
                B1b[j].y = w1[colb + 1];
                // A 16x4 row 0 = vec: broadcast LDS read + branchless select
                const int   ka = hi ? (k + 2) : k;
                const float vx = vec[ka];         // zero-padded past K
                const float vy = vec[ka + 1];
                Ab[j].x = (n == 0) ? vx : 0.0f;
                Ab[j].y = (n == 0) ? vy : 0.0f;
            }
        }

        // --- drain: 2 tiles x 2 K-parities = 4 independent chains ----------
#pragma unroll
        for (int j = 0; j < BATCH; ++j) {
            const int c = cb + j;
            if (c < NCH) {
                if ((c & 1) == 0) {
                    a00 = wmma4(Ab[j], B0b[j], a00);
                    a10 = wmma4(Ab[j], B1b[j], a10);
                } else {
                    a01 = wmma4(Ab[j], B0b[j], a01);
                    a11 = wmma4(Ab[j], B1b[j], a11);
                }
            }
        }
    }
}

__global__ void __launch_bounds__(NTHREADS)
lstm_step_kernel(const float* __restrict__ x,       // [100]
                 const float* __restrict__ h0,      // [2*150]
                 const float* __restrict__ c0,      // [2*150]
                 const float* __restrict__ W_ih0,   // [600*100]
                 const float* __restrict__ W_hh0,   // [600*150]
                 const float* __restrict__ b_ih0,   // [600]
                 const float* __restrict__ b_hh0,   // [600]
                 const float* __restrict__ W_ih1,   // [600*150]
                 const float* __restrict__ W_hh1,   // [600*150]
                 const float* __restrict__ b_ih1,   // [600]
                 const float* __restrict__ b_hh1,   // [600]
                 const float* __restrict__ W_lin,   // [150]
                 const float* __restrict__ b_lin,   // [1]
                 float* __restrict__ out)           // [1]
{
    __shared__ float sGates[NTILES * 16];  // 608 gate slots (600 used)
    __shared__ float sVec[160];            // layer input vector (x, then h1)
    __shared__ float sHprev[160];          // h_prev for current layer
    __shared__ float sH2[160];             // final hidden state

    const int tid  = threadIdx.x;
    const int lane = tid & 31;
    const int wave = tid >> 5;
    const int n    = lane & 15;

    // Per-wave tile assignment: tiles t0 = wave, t1 = wave + 19.
    const int t0 = wave;
    const int t1 = wave + NWAVES;
    int r0 = t0 * 16 + n;                       // <= 303, always valid
    int r1 = t1 * 16 + n;
    if (r1 > G4 - 1) r1 = G4 - 1;               // clamped rows never written back
    const int g0 = t0 * 16 + lane;              // write index (lane<16 only)
    const int g1 = t1 * 16 + lane;

    // ---- Phase A: stage vectors into LDS; prefetch all weights ------------
    if (tid < 160) {
        sVec[tid]   = (tid < LAG) ? x[tid]  : 0.0f;
        sHprev[tid] = (tid < HID) ? h0[tid] : 0.0f;
        sH2[tid]    = 0.0f;
    }
    {
        // Fire-and-forget cacheline prefetch (128B stride) of all weights.
        const int STRIDE = NTHREADS * 32;       // floats per sweep step
        for (int o = tid * 32; o < G4 * LAG; o += STRIDE)
            __builtin_prefetch(W_ih0 + o, 0, 3);
        for (int o = tid * 32; o < G4 * HID; o += STRIDE) {
            __builtin_prefetch(W_hh0 + o, 0, 3);
            __builtin_prefetch(W_ih1 + o, 0, 3);
            __builtin_prefetch(W_hh1 + o, 0, 3);
        }
    }
    __syncthreads();

    // ---- Phase B: layer-0 gates = W_ih0*x + W_hh0*h0[0] + biases ----------
    {
        const v8f z = {0.f, 0.f, 0.f, 0.f, 0.f, 0.f, 0.f, 0.f};
        v8f a00 = z, a01 = z, a10 = z, a11 = z;
        wmma_matvec2<LAG>(a00, a01, a10, a11, W_ih0, LAG, sVec,   r0, r1, lane);
        wmma_matvec2<HID>(a00, a01, a10, a11, W_hh0, HID, sHprev, r0, r1, lane);
        if (lane < 16) {
            sGates[g0] = a00[0] + a01[0] + b_ih0[g0] + b_hh0[g0];
            if (g1 < G4)
                sGates[g1] = a10[0] + a11[0] + b_ih0[g1] + b_hh0[g1];
        }
    }
    __syncthreads();

    // ---- Phase C: layer-0 activations -> h1 (reuse sVec); stage h0[1] -----
    if (tid < 160) {
        float h1v = 0.0f, hp = 0.0f;
        if (tid < HID) {
            const float i_ = sigmoidf_(sGates[tid]);
            const float f_ = sigmoidf_(sGates[HID + tid]);
            const float g_ = tanhf(sGates[2 * HID + tid]);
            const float o_ = sigmoidf_(sGates[3 * HID + tid]);
            const float c_ = f_ * c0[tid] + i_ * g_;
            h1v = o_ * tanhf(c_);
            hp  = h0[HID + tid];
        }
        sVec[tid]   = h1v;
        sHprev[tid] = hp;
    }
    __syncthreads();

    // ---- Phase D: layer-1 gates = W_ih1*h1 + W_hh1*h0[1] + biases ---------
    {
        const v8f z = {0.f, 0.f, 0.f, 0.f, 0.f, 0.f, 0.f, 0.f};
        v8f a00 = z, a01 = z, a10 = z, a11 = z;
        wmma_matvec2<HID>(a00, a01, a10, a11, W_ih1, HID, sVec,   r0, r1, lane);
        wmma_matvec2<HID>(a00, a01, a10, a11, W_hh1, HID, sHprev, r0, r1, lane);
        if (lane < 16) {
            sGates[g0] = a00[0] + a01[0] + b_ih1[g0] + b_hh1[g0];
            if (g1 < G4)
                sGates[g1] = a10[0] + a11[0] + b_ih1[g1] + b_hh1[g1];
        }
    }
    __syncthreads();

    // ---- Phase E: layer-1 activations -> h2 -------------------------------
    if (tid < HID) {
        const float i_ = sigmoidf_(sGates[tid]);
        const float f_ = sigmoidf_(sGates[HID + tid]);
        const float g_ = tanhf(sGates[2 * HID + tid]);
        const float o_ = sigmoidf_(sGates[3 * HID + tid]);
        const float c_ = f_ * c0[HID + tid] + i_ * g_;
        sH2[tid] = o_ * tanhf(c_);
    }
    __syncthreads();

    // ---- Phase F: linear head, wave-0 shuffle reduction -------------------
    if (wave == 0) {
        float s = 0.0f;
        for (int i = lane; i < HID; i += 32)
            s += sH2[i] * W_lin[i];
        for (int off = 16; off > 0; off >>= 1)
            s += __shfl_xor(s, off, 32);
        if (lane == 0)
            out[0] = s + b_lin[0];
    }
}

extern "C" void kernel_launch(void* const* d_in, const int* in_sizes, int n_in,
                              void* d_out, int out_size, void* d_ws, size_t ws_size,
                              hipStream_t stream) {
    (void)in_sizes; (void)n_in; (void)d_ws; (void)ws_size; (void)out_size;
    const float* x     = (const float*)d_in[0];
    const float* h0    = (const float*)d_in[1];
    const float* c0    = (const float*)d_in[2];
    const float* W_ih0 = (const float*)d_in[3];
    const float* W_hh0 = (const float*)d_in[4];
    const float* b_ih0 = (const float*)d_in[5];
    const float* b_hh0 = (const float*)d_in[6];
    const float* W_ih1 = (const float*)d_in[7];
    const float* W_hh1 = (const float*)d_in[8];
    const float* b_ih1 = (const float*)d_in[9];
    const float* b_hh1 = (const float*)d_in[10];
    const float* W_lin = (const float*)d_in[11];
    const float* b_lin = (const float*)d_in[12];
    float* out = (float*)d_out;

    lstm_step_kernel<<<1, NTHREADS, 0, stream>>>(
        x, h0, c0, W_ih0, W_hh0, b_ih0, b_hh0,
        W_ih1, W_hh1, b_ih1, b_hh1, W_lin, b_lin, out);
}